// SurrogatePrecision_79577154060375
// MI455X (gfx1250) — compile-verified
//
#include <hip/hip_runtime.h>
#include <hip/hip_bf16.h>

// Problem constants (fixed by the reference)
#define N 512      // queries / gallery
#define F 384      // feature dim
#define KCONST 6.0f
#define INV_T2 100.0f               // 1/0.01 ; T1 == 1.0
#define LOG2E  1.44269504088896340736f

typedef __attribute__((ext_vector_type(2))) float v2f;
typedef __attribute__((ext_vector_type(8))) float v8f;

// ---------------------------------------------------------------------------
// Pass 0: squared row norms sq[r] = sum_k feat[r][k]^2
// ---------------------------------------------------------------------------
__global__ __launch_bounds__(256) void sqnorm_kernel(const float* __restrict__ feat,
                                                     float* __restrict__ sq) {
    const int r = blockIdx.x * 256 + threadIdx.x;
    if (r >= N) return;
    const float4* row = (const float4*)(feat + r * F);
    float s = 0.0f;
#pragma unroll 4
    for (int k = 0; k < F / 4; ++k) {
        float4 v = row[k];
        s += v.x * v.x + v.y * v.y + v.z * v.z + v.w * v.w;
    }
    sq[r] = s;
}

// ---------------------------------------------------------------------------
// Pass 1: pairwise distances via WMMA f32 16x16x4.
// One wave (32 lanes) per 16x16 tile of D. C[m][n] = x_m . x_n (Gram), then
// d = sqrt(max(sq[m] + sq[n] - 2*gram, 0)) fused into the store.
//
// A (16x4 f32) fragment layout per ISA: lanes 0-15 -> M=lane, VGPR0=K0,VGPR1=K1;
// lanes 16-31 -> M=lane-16, VGPR0=K2,VGPR1=K3.  => each lane loads 2 consecutive
// floats at k + 2*(lane>=16).  B (4x16) mirrors this with N=lane%16, so since
// B[k][n] = feat[colBase+n][k], the load pattern is identical with the column base.
// ---------------------------------------------------------------------------
__global__ __launch_bounds__(32) void dist_kernel(const float* __restrict__ feat,
                                                  const float* __restrict__ sq,
                                                  float* __restrict__ D) {
    const int tileM = blockIdx.x * 16;
    const int tileN = blockIdx.y * 16;
    const int lane  = threadIdx.x;        // 0..31
    const int half  = lane >> 4;          // 0: K pair (0,1), 1: K pair (2,3)
    const int lm    = lane & 15;

    const float* aBase = feat + (tileM + lm) * F + 2 * half;
    const float* bBase = feat + (tileN + lm) * F + 2 * half;

    v8f acc = {};
#pragma unroll 8
    for (int k = 0; k < F; k += 4) {
        v2f a = *(const v2f*)(aBase + k);   // global_load_b64
        v2f b = *(const v2f*)(bBase + k);   // global_load_b64
        // 8 args: (neg_a, A, neg_b, B, c_mod, C, reuse_a, reuse_b)
        acc = __builtin_amdgcn_wmma_f32_16x16x4_f32(
            false, a, false, b, (short)0, acc, false, false);
    }

    // C/D layout: VGPR i -> M = i (lanes 0-15) or i+8 (lanes 16-31); N = lane%16.
    const float sqN = sq[tileN + lm];
#pragma unroll
    for (int i = 0; i < 8; ++i) {
        const int m = i + 8 * half;
        const float d2 = sq[tileM + m] + sqN - 2.0f * acc[i];
        // raw v_sqrt_f32 (skip IEEE range-fixup expansion)
        D[(tileM + m) * N + (tileN + lm)] =
            __builtin_amdgcn_sqrtf(fmaxf(d2, 0.0f));
    }
}

// ---------------------------------------------------------------------------
// Pass 2: per-query rank + masked precision.  One block (256 thr = 8 waves)
// per query q.  D[q,:] staged in LDS pre-scaled by C = INV_T2*log2(e), so each
// sigmoid is:  sub -> v_exp_f32 -> add1 -> v_rcp_f32 -> acc  (3 VALU + 2 TRANS).
//   sigmoid((dg-dp)*INV_T2) = 1/(1 + exp2(dp*C - dg*C))
// LDS row read as float4 (ds_load_b128 broadcast, no bank conflicts).
// ---------------------------------------------------------------------------
__global__ __launch_bounds__(256) void rank_kernel(const float* __restrict__ D,
                                                   const int* __restrict__ labels,
                                                   float* __restrict__ prec) {
    __shared__ float drow[N];     // distances pre-scaled by C
    __shared__ float red[256];

    const int q   = blockIdx.x;
    const int tid = threadIdx.x;
    const float C = INV_T2 * LOG2E;

    for (int i = tid; i < N; i += 256) drow[i] = D[q * N + i] * C;
    __syncthreads();

    const int myLab = labels[q];
    float psum = 0.0f;

#pragma unroll
    for (int gi = 0; gi < 2; ++gi) {
        const int g = tid + 256 * gi;
        const float dg = drow[g];          // already scaled
        float rank = 0.0f;
        const float4* dr4 = (const float4*)drow;
#pragma unroll 4
        for (int gp = 0; gp < N / 4; ++gp) {
            float4 v = dr4[gp];
            rank += __builtin_amdgcn_rcpf(1.0f + __builtin_amdgcn_exp2f(v.x - dg));
            rank += __builtin_amdgcn_rcpf(1.0f + __builtin_amdgcn_exp2f(v.y - dg));
            rank += __builtin_amdgcn_rcpf(1.0f + __builtin_amdgcn_exp2f(v.z - dg));
            rank += __builtin_amdgcn_rcpf(1.0f + __builtin_amdgcn_exp2f(v.w - dg));
        }
        // pg = sigmoid(KCONST - rank) = 1/(1 + exp2((rank-K)*log2e))
        const float pg =
            __builtin_amdgcn_rcpf(1.0f + __builtin_amdgcn_exp2f((rank - KCONST) * LOG2E));
        psum += (labels[g] == myLab) ? pg : 0.0f;
    }

    red[tid] = psum;
    __syncthreads();
#pragma unroll
    for (int s = 128; s > 0; s >>= 1) {
        if (tid < s) red[tid] += red[tid + s];
        __syncthreads();
    }
    if (tid == 0) prec[q] = red[0] * (1.0f / KCONST);
}

// ---------------------------------------------------------------------------
// Pass 3: out = 1 - mean(prec)
// ---------------------------------------------------------------------------
__global__ __launch_bounds__(256) void finalize_kernel(const float* __restrict__ prec,
                                                       float* __restrict__ out) {
    __shared__ float red[256];
    const int tid = threadIdx.x;
    red[tid] = prec[tid] + prec[tid + 256];
    __syncthreads();
#pragma unroll
    for (int s = 128; s > 0; s >>= 1) {
        if (tid < s) red[tid] += red[tid + s];
        __syncthreads();
    }
    if (tid == 0) out[0] = 1.0f - red[0] * (1.0f / (float)N);
}

// ---------------------------------------------------------------------------
extern "C" void kernel_launch(void* const* d_in, const int* in_sizes, int n_in,
                              void* d_out, int out_size, void* d_ws, size_t ws_size,
                              hipStream_t stream) {
    (void)in_sizes; (void)n_in; (void)out_size; (void)ws_size;

    const float* feat   = (const float*)d_in[0];   // [512,384] f32
    const int*   labels = (const int*)d_in[1];     // [512] i32
    float*       out    = (float*)d_out;           // scalar f32

    // Workspace layout: sq[512] | prec[512] | pad | D[512*512]
    float* sq   = (float*)d_ws;
    float* prec = sq + N;
    float* D    = (float*)((char*)d_ws + 4096);

    sqnorm_kernel<<<dim3(N / 256), dim3(256), 0, stream>>>(feat, sq);

    dim3 dgrid(N / 16, N / 16);
    dist_kernel<<<dgrid, dim3(32), 0, stream>>>(feat, sq, D);

    rank_kernel<<<dim3(N), dim3(256), 0, stream>>>(D, labels, prec);

    finalize_kernel<<<dim3(1), dim3(256), 0, stream>>>(prec, out);
}